// ResidualGCNModel_11510512353283
// MI455X (gfx1250) — compile-verified
//
#include <hip/hip_runtime.h>

// ---------------------------------------------------------------------------
// ResidualGCN on MI455X (gfx1250): fp32 WMMA GEMMs + atomic scatter aggregation
// ---------------------------------------------------------------------------

typedef __attribute__((ext_vector_type(2))) float v2f;
typedef __attribute__((ext_vector_type(4))) float v4f;
typedef __attribute__((ext_vector_type(8))) float v8f;

#define NN 50000
#define NE 640000
#define KD 128   // fan_in of every GEMM (IN_CH == HID_CH == 128)

__device__ __forceinline__ void atomic_add_f32(float* p, float v) {
    __hip_atomic_fetch_add(p, v, __ATOMIC_RELAXED, __HIP_MEMORY_SCOPE_AGENT);
}

// ---------------- degree / normalization ----------------------------------

__global__ void k_deg_init(float* __restrict__ deg) {
    int i = blockIdx.x * blockDim.x + threadIdx.x;
    if (i < NN) deg[i] = 1.0f;                 // self-loop
}

__global__ void k_deg_count(const int* __restrict__ ei, float* __restrict__ deg) {
    int e = blockIdx.x * blockDim.x + threadIdx.x;
    if (e < NE) atomic_add_f32(&deg[ei[NE + e]], 1.0f);   // col side
}

__global__ void k_deg_finalize(float* __restrict__ deg) {
    int i = blockIdx.x * blockDim.x + threadIdx.x;
    if (i < NN) deg[i] = rsqrtf(deg[i]);       // deg >= 1 always
}

// ---------------- fp32 WMMA GEMM: C[M x N] = act(A + bin) @ W (+ b0 + b1) --
// A: M x 128 row-major. W: 128 x N row-major, staged into LDS in *fragment
// order*: K-pair p, column n -> ldsW[(p*N + n)*2 + {0,1}] = W[2p..2p+1][n],
// so a lane's B-frag is a single aligned ds_load_b64 (no repack movs).
// One wave owns a 16-row strip and all N columns (N/16 fp32 16x16 frags).
// RELU (compile-time): A element k passes through relu(A + bias_in[k]).

template<int N, bool RELU>
__global__ __launch_bounds__(256)
void k_gemm_wmma(const float* __restrict__ A, const float* __restrict__ W,
                 const float* __restrict__ bias_in,
                 const float* __restrict__ bias_out0,
                 const float* __restrict__ bias_out1,
                 float* __restrict__ C, int M)
{
    __shared__ float ldsW[KD * N];           // swizzled pairs
    __shared__ float ldsBin[KD];

    const int tid = threadIdx.x;
    {   // stage W into LDS, swizzled to fragment order
        constexpr int PAIRS = (KD / 2) * N;
        v2f* Lv = (v2f*)ldsW;
        for (int i = tid; i < PAIRS; i += 256) {
            const int p = i / N;             // K-pair index (k = 2p, 2p+1)
            const int n = i - p * N;
            v2f w;
            w[0] = W[(2 * p)     * N + n];
            w[1] = W[(2 * p + 1) * N + n];
            Lv[i] = w;
        }
        if (RELU) { if (tid < KD) ldsBin[tid] = bias_in[tid]; }
    }
    __syncthreads();

    const int wave = tid >> 5;
    const int lane = tid & 31;
    const int m0   = blockIdx.x * 128 + wave * 16;
    if (m0 >= M) return;                     // wave-uniform: EXEC stays full

    const int l15 = lane & 15;
    const int hh  = lane >> 4;               // which K-pair half this lane holds
    constexpr int NT = N / 16;

    v8f acc[NT] = {};
    const v2f* Arow2 = (const v2f*)(A + (size_t)(m0 + l15) * KD);
    const v2f* Lb    = (const v2f*)ldsW;

#pragma unroll 2
    for (int kk = 0; kk < KD; kk += 4) {
        const int kb = kk + 2 * hh;          // even; this lane's K base
        // A frag: 16x4, lanes 0-15 hold K={kk,kk+1}, lanes 16-31 K={kk+2,kk+3}
        v2f a = Arow2[kb >> 1];              // one global_load_b64
        if (RELU) {
            a[0] = fmaxf(a[0] + ldsBin[kb],     0.0f);
            a[1] = fmaxf(a[1] + ldsBin[kb + 1], 0.0f);
        }
        const int p = kb >> 1;               // swizzled K-pair row in LDS
#pragma unroll
        for (int t = 0; t < NT; ++t) {
            const int n = t * 16 + l15;
            v2f b = Lb[p * N + n];           // one aligned ds_load_b64
            acc[t] = __builtin_amdgcn_wmma_f32_16x16x4_f32(
                false, a, false, b, (short)0, acc[t], false, false);
        }
    }

#pragma unroll
    for (int t = 0; t < NT; ++t) {
        const int n = t * 16 + l15;
        float badd = 0.0f;
        if (bias_out0) badd += bias_out0[n];
        if (bias_out1) badd += bias_out1[n];
#pragma unroll
        for (int r = 0; r < 8; ++r) {
            const int row = m0 + r + 8 * hh; // C/D layout: M = r + 8*(lane>=16)
            C[(size_t)row * N + n] = acc[t][r] + badd;
        }
    }
}

// ---------------- aggregation ----------------------------------------------
// self-loop term: out[i] (+)= dinv[i]^2 * t[i]       (vectorized, no atomics)

template<int C, bool ACCUM>
__global__ void k_agg_self(const float* __restrict__ t, const float* __restrict__ dinv,
                           float* __restrict__ out)
{
    constexpr int PV = C / 4;                // float4 per node
    int gid = blockIdx.x * blockDim.x + threadIdx.x;
    if (gid >= NN * PV) return;
    int node = gid / PV;
    float d = dinv[node];
    float s = d * d;
    v4f v = ((const v4f*)t)[gid] * s;
    if (ACCUM) v += ((const v4f*)out)[gid];
    ((v4f*)out)[gid] = v;
}

// edge scatter: out[col] += dinv[row]*dinv[col] * t[row]; one wave per edge,
// lanes stripe the channels (float4 loads, per-dword f32 atomics).

template<int C>
__global__ __launch_bounds__(256)
void k_agg_edges(const int* __restrict__ ei, const float* __restrict__ dinv,
                 const float* __restrict__ t, float* __restrict__ out)
{
    constexpr int VPL = C / 32;              // floats per lane (4 or 2)
    const int lane = threadIdx.x & 31;
    const int wid  = (blockIdx.x * blockDim.x + threadIdx.x) >> 5;
    const int nw   = (gridDim.x * blockDim.x) >> 5;

    for (int e = wid; e < NE; e += nw) {
        const int r = ei[e];
        const int c = ei[NE + e];
        const float nrm = dinv[r] * dinv[c];
        const float* src = t   + (size_t)r * C + lane * VPL;
        float*       dst = out + (size_t)c * C + lane * VPL;
        float vals[VPL];
#pragma unroll
        for (int j = 0; j < VPL; ++j) vals[j] = src[j];
#pragma unroll
        for (int j = 0; j < VPL; ++j) atomic_add_f32(dst + j, nrm * vals[j]);
    }
}

// ---------------- launcher --------------------------------------------------

extern "C" void kernel_launch(void* const* d_in, const int* in_sizes, int n_in,
                              void* d_out, int out_size, void* d_ws, size_t ws_size,
                              hipStream_t stream) {
    const float* x    = (const float*)d_in[0];   // 50000 x 128
    const int*   ei   = (const int*)d_in[1];     // 2 x 640000
    const float* W1   = (const float*)d_in[2];
    const float* b1   = (const float*)d_in[3];
    const float* W2   = (const float*)d_in[4];
    const float* b2   = (const float*)d_in[5];
    const float* W3   = (const float*)d_in[6];   // 128 x 64
    const float* b3   = (const float*)d_in[7];
    const float* Wres = (const float*)d_in[8];   // 128 x 64
    const float* bres = (const float*)d_in[9];
    float* out = (float*)d_out;                  // 50000 x 64

    char* ws = (char*)d_ws;
    float* dinv = (float*)ws;                              // 50000 f32
    float* bufA = (float*)(ws + 200192);                   // 50000 x 128 (16B aligned)
    float* bufB = bufA + (size_t)NN * KD;                  // 50000 x 128

    const int TB = 256;
    const int gN  = (NN + TB - 1) / TB;
    const int gE  = (NE + TB - 1) / TB;
    const int gMM = (NN + 127) / 128;                      // GEMM row blocks (391)
    const int gE8 = 20000;                                 // 160k waves, 4 edges each

    // 1) symmetric normalization: dinv = rsqrt(1 + indegree)
    k_deg_init    <<<gN, TB, 0, stream>>>(dinv);
    k_deg_count   <<<gE, TB, 0, stream>>>(ei, dinv);
    k_deg_finalize<<<gN, TB, 0, stream>>>(dinv);

    // 2) layer 1: t1 = x @ W1 ; agg -> bufB
    k_gemm_wmma<128, false><<<gMM, TB, 0, stream>>>(x, W1, nullptr, nullptr, nullptr, bufA, NN);
    k_agg_self<128, false><<<(NN * 32 + TB - 1) / TB, TB, 0, stream>>>(bufA, dinv, bufB);
    k_agg_edges<128><<<gE8, TB, 0, stream>>>(ei, dinv, bufA, bufB);

    // 3) layer 2: t2 = relu(bufB + b1) @ W2 ; agg -> bufB
    k_gemm_wmma<128, true><<<gMM, TB, 0, stream>>>(bufB, W2, b1, nullptr, nullptr, bufA, NN);
    k_agg_self<128, false><<<(NN * 32 + TB - 1) / TB, TB, 0, stream>>>(bufA, dinv, bufB);
    k_agg_edges<128><<<gE8, TB, 0, stream>>>(ei, dinv, bufA, bufB);

    // 4) layer 3 transform: t3 = relu(bufB + b2) @ W3 -> bufA (50000 x 64)
    k_gemm_wmma<64, true><<<gMM, TB, 0, stream>>>(bufB, W3, b2, nullptr, nullptr, bufA, NN);

    // 5) residual straight into d_out: out = x @ Wres + bres + b3
    k_gemm_wmma<64, false><<<gMM, TB, 0, stream>>>(x, Wres, nullptr, bres, b3, out, NN);

    // 6) layer-3 aggregation accumulates into d_out (stream-ordered)
    k_agg_self<64, true><<<(NN * 16 + TB - 1) / TB, TB, 0, stream>>>(bufA, dinv, out);
    k_agg_edges<64><<<gE8, TB, 0, stream>>>(ei, dinv, bufA, out);
}